// DetectionLayer_86260123173892
// MI455X (gfx1250) — compile-verified
//
#include <hip/hip_runtime.h>
#include <stdint.h>

// YOLO detection layer: (16, 255, 76, 76) f32 -> (16, 17328, 85) f32
// Pure bandwidth-bound transpose + elementwise (sigmoid/exp). No matmul -> no WMMA.
// CDNA5 path used: GLOBAL_LOAD_ASYNC_TO_LDS_B128 (ASYNCcnt) for the global->LDS
// staging of the transpose, then coalesced dword stores.

#define NG      76
#define NGSQ    (NG * NG)          // 5776  (divisible by 4 -> b128 chunks ok)
#define NCH     85                 // 5 + 80 classes
#define NA      3
#define TG      128                // g-values per tile
#define PITCHF  132                // LDS row pitch in floats (528 B, 16B multiple)
#define NTILES  ((NGSQ + TG - 1) / TG)   // 46

__launch_bounds__(256)
__global__ void detection_layer_kernel(const float* __restrict__ x,
                                       float* __restrict__ out) {
    __shared__ __align__(16) float tile[NCH * PITCHF];   // 44,880 B

    const int tid  = threadIdx.x;
    const int lane = tid & 31;
    const int wave = tid >> 5;                // 0..7
    const int tIdx = blockIdx.x;              // tile within slab
    const int slab = blockIdx.y;              // b*3 + a
    const int a    = slab % NA;

    const int gBase  = tIdx * TG;
    const int validG = min(TG, NGSQ - gBase);

    // slab base: x + (b*255 + a*85) * 5776
    const float* src = x + (size_t)slab * (size_t)(NCH * NGSQ);

    // ---------------- Phase 1: async global -> LDS (transpose staging) -------
    // Each wave handles channels c = wave, wave+8, ... ; per channel one
    // global_load_async_to_lds_b128: 32 lanes x 16B = 512B = 128 floats of g.
    uint32_t ldsBase = (uint32_t)(uintptr_t)(&tile[0]);   // low 32 bits = LDS byte offset

    int gChunk = gBase + lane * 4;                         // first g of this lane's 16B
    if (gChunk > NGSQ - 4) gChunk = NGSQ - 4;              // clamp: stay inside channel row

    for (int c = wave; c < NCH; c += 8) {
        uint32_t ldsOff = ldsBase + (uint32_t)(c * (PITCHF * 4)) + (uint32_t)(lane * 16);
        uint32_t gOff   = (uint32_t)((c * NGSQ + gChunk) * 4);
        asm volatile("global_load_async_to_lds_b128 %0, %1, %2"
                     :: "v"(ldsOff), "v"(gOff), "s"(src)
                     : "memory");
    }
    asm volatile("s_wait_asynccnt 0x0" ::: "memory");
    __syncthreads();

    // ---------------- Phase 2: LDS -> math -> coalesced contiguous stores ----
    // exp(w) * (anchor/stride) * stride == exp(w) * anchor  (stride cancels)
    const float aw = (a == 0) ? 10.0f : ((a == 1) ? 16.0f : 33.0f);
    const float ah = (a == 0) ? 13.0f : ((a == 1) ? 30.0f : 23.0f);
    const float strideF = 8.0f;    // 608 / 76

    float* dst = out + (size_t)slab * (size_t)(NGSQ * NCH) + (size_t)gBase * NCH;
    const int total = validG * NCH;                         // <= 10880

#pragma unroll 1
    for (int j = 0; j < (TG * NCH + 255) / 256; ++j) {      // 43 iterations
        int idx = tid + j * 256;
        if (idx < total) {
            uint32_t g = (uint32_t)idx / NCH;               // g within tile
            uint32_t c = (uint32_t)idx - g * NCH;           // channel 0..84
            float v = tile[c * PITCHF + g];

            uint32_t gg = (uint32_t)gBase + g;              // 0..5775
            uint32_t gy = gg / NG;
            uint32_t gx = gg - gy * NG;

            bool iswh = (c == 2) || (c == 3);
            float t = __expf(iswh ? v : -v);                // exp(v) for w/h, exp(-v) for sigmoid
            float s = 1.0f / (1.0f + t);                    // sigmoid(v) when !iswh

            float r;
            if (c < 2) {
                r = (s + (float)(c == 0 ? gx : gy)) * strideF;
            } else if (iswh) {
                r = t * ((c == 2) ? aw : ah);
            } else {
                r = s;                                      // conf / class probs
            }
            dst[idx] = r;                                   // fully coalesced
        }
    }
}

extern "C" void kernel_launch(void* const* d_in, const int* in_sizes, int n_in,
                              void* d_out, int out_size, void* d_ws, size_t ws_size,
                              hipStream_t stream) {
    const float* x = (const float*)d_in[0];
    float* out = (float*)d_out;

    // nB derived from input size: nB * 255 * 5776 elements
    int nB = in_sizes[0] / (NA * NCH * NGSQ);
    dim3 grid(NTILES, nB * NA);   // 46 x 48
    detection_layer_kernel<<<grid, 256, 0, stream>>>(x, out);
}